// HAN_45260365365442
// MI455X (gfx1250) — compile-verified
//
#include <hip/hip_runtime.h>
#include <hip/hip_bf16.h>
#include <math.h>

// ---------------- CDNA5 WMMA types ----------------
typedef __attribute__((ext_vector_type(2))) float v2f;
typedef __attribute__((ext_vector_type(8))) float v8f;

#define NEG_SLOPE 0.2f
#define GEMM_MT 5   // M-tiles per wave; 50000/16=3125 and 10000/16=625 are divisible by 5

// ---------------- utility kernels ----------------
__global__ void k_fill_f32(float* __restrict__ p, float v, int n) {
  int i = blockIdx.x * blockDim.x + threadIdx.x;
  if (i < n) p[i] = v;
}
__global__ void k_fill_u32(unsigned* __restrict__ p, unsigned v, int n) {
  int i = blockIdx.x * blockDim.x + threadIdx.x;
  if (i < n) p[i] = v;
}
__global__ void k_relu(float* __restrict__ p, int n) {
  int i = blockIdx.x * blockDim.x + threadIdx.x;
  if (i < n) p[i] = fmaxf(p[i], 0.f);
}

// ---------------- WMMA GEMM: out[N,F] = X[N,K] @ W[K,F] + bias[F] ----------------
// One wave per (GEMM_MT x 16)-row by 16-col output strip; V_WMMA_F32_16X16X4_F32.
// Each B fragment (4x16) is reused across GEMM_MT row tiles.
// A (16x4 f32): lane L<16 -> row M=L, K=k+{0,1}; lane>=16 -> row M=L-16, K=k+{2,3}.
// B (4x16 f32): lane L -> col N=L&15, same K split.
// C/D (16x16 f32): vgpr i, lanes 0-15 -> M=i, N=lane; lanes 16-31 -> M=i+8, N=lane-16.
__global__ void k_gemm_wmma_bias(const float* __restrict__ X, const float* __restrict__ Wm,
                                 const float* __restrict__ bias, float* __restrict__ out,
                                 int N, int K, int F) {
  int wave = (int)((blockIdx.x * blockDim.x + threadIdx.x) >> 5);
  int lane = threadIdx.x & 31;
  int tiles_f = F >> 4;
  int strips_m = (N >> 4) / GEMM_MT;          // N assumed divisible by 16*GEMM_MT
  if (wave >= strips_m * tiles_f) return;     // wave-uniform: EXEC all-ones inside
  int tm5 = wave / tiles_f, tf = wave - tm5 * tiles_f;
  int f = (tf << 4) + (lane & 15);
  int koff = (lane >> 4) << 1;
  int mbase = tm5 * (GEMM_MT << 4) + (lane & 15);
  const float* xr0 = X + (size_t)mbase * K;
  v8f acc[GEMM_MT];
#pragma unroll
  for (int mt = 0; mt < GEMM_MT; ++mt) acc[mt] = (v8f){};
  for (int k = 0; k < K; k += 4) {
    v2f b;
    b.x = Wm[(size_t)(k + koff) * F + f];
    b.y = Wm[(size_t)(k + koff + 1) * F + f];
#pragma unroll
    for (int mt = 0; mt < GEMM_MT; ++mt) {
      const float* xr = xr0 + ((size_t)(mt << 4)) * K;
      v2f a;
      a.x = xr[k + koff];
      a.y = xr[k + koff + 1];
      acc[mt] = __builtin_amdgcn_wmma_f32_16x16x4_f32(false, a, false, b, (short)0, acc[mt],
                                                      false, false);
    }
  }
  float bv = bias[f];
  int rowh = (lane >> 4) << 3;
#pragma unroll
  for (int mt = 0; mt < GEMM_MT; ++mt) {
    int row0 = tm5 * (GEMM_MT << 4) + (mt << 4) + rowh;
#pragma unroll
    for (int i = 0; i < 8; ++i)
      out[(size_t)(row0 + i) * F + f] = acc[mt][i] + bv;
  }
}

// ---------------- fused semantic score: *wout += sum_n q . tanh(Z[n]@kW + kb) ----------------
__global__ void k_semantic_score(const float* __restrict__ Z, const float* __restrict__ kW,
                                 const float* __restrict__ kb, const float* __restrict__ q,
                                 float* __restrict__ wout, int N, int F) {
  int wave = (int)((blockIdx.x * blockDim.x + threadIdx.x) >> 5);
  int lane = threadIdx.x & 31;
  int tiles_f = F >> 4;
  int tiles_m = N >> 4;
  if (wave >= tiles_m * tiles_f) return;
  int tm = wave / tiles_f, tf = wave - tm * tiles_f;
  int m = (tm << 4) + (lane & 15);
  int f = (tf << 4) + (lane & 15);
  int koff = (lane >> 4) << 1;
  const float* zr = Z + (size_t)m * F;
  v8f acc = {};
  for (int k = 0; k < F; k += 4) {
    v2f a, b;
    a.x = zr[k + koff];
    a.y = zr[k + koff + 1];
    b.x = kW[(size_t)(k + koff) * F + f];
    b.y = kW[(size_t)(k + koff + 1) * F + f];
    acc = __builtin_amdgcn_wmma_f32_16x16x4_f32(false, a, false, b, (short)0, acc, false, false);
  }
  float kbf = kb[f], qf = q[f];
  float partial = 0.f;
#pragma unroll
  for (int i = 0; i < 8; ++i) partial += tanhf(acc[i] + kbf) * qf;
  for (int off = 16; off > 0; off >>= 1) partial += __shfl_down(partial, off, 32);
  if (lane == 0) atomicAdd(wout, partial);
}

// ---------------- per-node attention dot: out[n,h] = sum_d h[n,h,d]*avec[h,d] ----------------
__global__ void k_att_node(const float* __restrict__ h, const float* __restrict__ avec,
                           float* __restrict__ out, int N, int H, int D) {
  int t = blockIdx.x * blockDim.x + threadIdx.x;
  if (t >= N * H) return;
  int n = t / H, hh = t - n * H;
  const float4* hp = (const float4*)(h + (size_t)n * H * D + (size_t)hh * D);
  const float4* ap = (const float4*)(avec + (size_t)hh * D);
  float s = 0.f;
  int d4 = D >> 2;
  for (int d = 0; d < d4; ++d) {
    float4 hv = hp[d], av = ap[d];
    s += hv.x * av.x + hv.y * av.y + hv.z * av.z + hv.w * av.w;
  }
  out[t] = s;
}

// ---------------- edge pass A: a = leaky_relu(as[src]+ad[dst]); segment-max via ordered-uint atomicMax
__global__ void k_edge_a(const int* __restrict__ src, const int* __restrict__ dst,
                         const float* __restrict__ as_, const float* __restrict__ ad_,
                         unsigned* __restrict__ m_enc, float* __restrict__ a_out,
                         int E, int H) {
  int t = blockIdx.x * blockDim.x + threadIdx.x;
  if (t >= E * H) return;
  int e = t / H, hh = t - e * H;
  float a = as_[(size_t)src[e] * H + hh] + ad_[(size_t)dst[e] * H + hh];
  a = a > 0.f ? a : NEG_SLOPE * a;
  a_out[t] = a;
  unsigned bits = __float_as_uint(a);
  unsigned enc = (bits & 0x80000000u) ? ~bits : (bits | 0x80000000u);
  atomicMax(&m_enc[(size_t)dst[e] * H + hh], enc);
}

// ---------------- edge pass B: e = exp(a - m[dst]); denom[dst] += e (in-place on a_out)
__global__ void k_edge_b(const int* __restrict__ dst, float* __restrict__ e_att,
                         const unsigned* __restrict__ m_enc, float* __restrict__ denom,
                         int E, int H) {
  int t = blockIdx.x * blockDim.x + threadIdx.x;
  if (t >= E * H) return;
  int e = t / H, hh = t - e * H;
  unsigned enc = m_enc[(size_t)dst[e] * H + hh];
  unsigned bits = (enc & 0x80000000u) ? (enc & 0x7FFFFFFFu) : ~enc;
  float mm = __uint_as_float(bits);
  float ex = __expf(e_att[t] - mm);
  e_att[t] = ex;
  atomicAdd(&denom[(size_t)dst[e] * H + hh], ex);
}

// ---------------- edge pass C (x4 vectorized): out[dst,f..f+3] += h_src[src,f..f+3] * alpha
__global__ void k_edge_c4(const int* __restrict__ src, const int* __restrict__ dst,
                          const float* __restrict__ e_att, const float* __restrict__ denom,
                          const float* __restrict__ hsrc, float* __restrict__ out,
                          int E, int H, int D) {
  int F = H * D;
  int Fq = F >> 2;                       // D divisible by 4 -> quad never straddles a head
  long long t = (long long)blockIdx.x * blockDim.x + threadIdx.x;
  if (t >= (long long)E * Fq) return;
  int e = (int)(t / Fq);
  int f4 = (int)(t - (long long)e * Fq);
  int f = f4 << 2;
  int hh = f / D;
  int de = dst[e], se = src[e];
  float alpha = e_att[(size_t)e * H + hh] / (denom[(size_t)de * H + hh] + 1e-16f);
  float4 hv = *(const float4*)(hsrc + (size_t)se * F + f);
  float* op = out + (size_t)de * F + f;
  atomicAdd(op + 0, hv.x * alpha);
  atomicAdd(op + 1, hv.y * alpha);
  atomicAdd(op + 2, hv.z * alpha);
  atomicAdd(op + 3, hv.w * alpha);
}

// ---------------- semantic combine for 2 relations (x4 vectorized) ----------------
__global__ void k_combine2(const float4* __restrict__ z0, const float4* __restrict__ z1,
                           const float* __restrict__ wsum, float inv_n,
                           float4* __restrict__ out, int n4) {
  int i = blockIdx.x * blockDim.x + threadIdx.x;
  if (i >= n4) return;
  float w0 = wsum[0] * inv_n, w1 = wsum[1] * inv_n;
  float m = fmaxf(w0, w1);
  float e0 = __expf(w0 - m), e1 = __expf(w1 - m);
  float inv_s = 1.f / (e0 + e1);
  e0 *= inv_s; e1 *= inv_s;
  float4 a = z0[i], b = z1[i], r;
  r.x = e0 * a.x + e1 * b.x;
  r.y = e0 * a.y + e1 * b.y;
  r.z = e0 * a.z + e1 * b.z;
  r.w = e0 * a.w + e1 * b.w;
  out[i] = r;
}

// ================= host-side orchestration =================
static inline int cdiv(long long a, int b) { return (int)((a + (long long)b - 1) / b); }

struct ETSpec {
  const int* src; const int* dst; int E; int Ns; int Nd;
  const float* hs; const float* hd; const float* avs; const float* avd;
  float* out; bool run;
};

static void hanconv_layer(const float* x_p, const float* x_t, int K, int H, int D,
                          const int* spp, const int* dpp, const int* spt, const int* dpt,
                          const int* stp, const int* dtp,
                          const float* Wp, const float* Wt, const float* bp, const float* bt,
                          const float* as_pp, const float* as_pt, const float* as_tp,
                          const float* ad_pp, const float* ad_pt, const float* ad_tp,
                          const float* kW, const float* kb, const float* q,
                          float* h_p, float* h_t, float* o_pp, float* o_pt, float* o_tp,
                          float* as_b, float* ad_b, unsigned* m_b, float* den_b, float* e_att,
                          float* wsum, float* out_p, bool compute_t, hipStream_t stream) {
  const int NP = 50000, NT = 10000;
  const int F = H * D;
  // 1) node projections via WMMA GEMM (5 row-tiles per wave)
  {
    int waves = ((NP / 16) / GEMM_MT) * (F / 16);
    k_gemm_wmma_bias<<<cdiv((long long)waves * 32, 256), 256, 0, stream>>>(x_p, Wp, bp, h_p, NP, K, F);
  }
  {
    int waves = ((NT / 16) / GEMM_MT) * (F / 16);
    k_gemm_wmma_bias<<<cdiv((long long)waves * 32, 256), 256, 0, stream>>>(x_t, Wt, bt, h_t, NT, K, F);
  }
  // 2) zero segment outputs
  k_fill_f32<<<cdiv((long long)NP * F, 256), 256, 0, stream>>>(o_pp, 0.f, NP * F);
  if (compute_t)
    k_fill_f32<<<cdiv((long long)NT * F, 256), 256, 0, stream>>>(o_pt, 0.f, NT * F);
  k_fill_f32<<<cdiv((long long)NP * F, 256), 256, 0, stream>>>(o_tp, 0.f, NP * F);

  ETSpec ets[3] = {
    { spp, dpp, 300000, NP, NP, h_p, h_p, as_pp, ad_pp, o_pp, true },
    { spt, dpt, 150000, NP, NT, h_p, h_t, as_pt, ad_pt, o_pt, compute_t },
    { stp, dtp, 150000, NT, NP, h_t, h_p, as_tp, ad_tp, o_tp, true },
  };
  for (int i = 0; i < 3; ++i) {
    ETSpec& et = ets[i];
    if (!et.run) continue;
    k_att_node<<<cdiv((long long)et.Ns * H, 256), 256, 0, stream>>>(et.hs, et.avs, as_b, et.Ns, H, D);
    k_att_node<<<cdiv((long long)et.Nd * H, 256), 256, 0, stream>>>(et.hd, et.avd, ad_b, et.Nd, H, D);
    // encoded -inf: ~bits(-inf) = ~0xFF800000 = 0x007FFFFF
    k_fill_u32<<<cdiv((long long)et.Nd * H, 256), 256, 0, stream>>>(m_b, 0x007FFFFFu, et.Nd * H);
    k_fill_f32<<<cdiv((long long)et.Nd * H, 256), 256, 0, stream>>>(den_b, 0.f, et.Nd * H);
    k_edge_a<<<cdiv((long long)et.E * H, 256), 256, 0, stream>>>(et.src, et.dst, as_b, ad_b, m_b, e_att, et.E, H);
    k_edge_b<<<cdiv((long long)et.E * H, 256), 256, 0, stream>>>(et.dst, e_att, m_b, den_b, et.E, H);
    k_edge_c4<<<cdiv((long long)et.E * (F >> 2), 256), 256, 0, stream>>>(et.src, et.dst, e_att, den_b,
                                                                         et.hs, et.out, et.E, H, D);
    k_relu<<<cdiv((long long)et.Nd * F, 256), 256, 0, stream>>>(et.out, et.Nd * F);
  }
  // 3) semantic attention for dst type "p" (relations: pp then tp, per EDGE_TYPES order)
  k_fill_f32<<<1, 32, 0, stream>>>(wsum, 0.f, 2);
  {
    int waves = (NP / 16) * (F / 16);
    k_semantic_score<<<cdiv((long long)waves * 32, 256), 256, 0, stream>>>(o_pp, kW, kb, q, wsum + 0, NP, F);
    k_semantic_score<<<cdiv((long long)waves * 32, 256), 256, 0, stream>>>(o_tp, kW, kb, q, wsum + 1, NP, F);
  }
  k_combine2<<<cdiv((long long)NP * (F >> 2), 256), 256, 0, stream>>>(
      (const float4*)o_pp, (const float4*)o_tp, wsum, 1.f / (float)NP, (float4*)out_p, NP * (F >> 2));
  // dst type "t": single relation -> softmax(beta)=1 -> h_t_out is o_pt (already ReLU'd in place)
}

extern "C" void kernel_launch(void* const* d_in, const int* in_sizes, int n_in,
                              void* d_out, int out_size, void* d_ws, size_t ws_size,
                              hipStream_t stream) {
  (void)in_sizes; (void)n_in; (void)out_size; (void)ws_size;
  const int NP = 50000, NT = 10000;
  const int F1 = 256, F2 = 64;

  const float* x_p = (const float*)d_in[0];
  const float* x_t = (const float*)d_in[1];
  const int* spp = (const int*)d_in[2];
  const int* dpp = (const int*)d_in[3];
  const int* spt = (const int*)d_in[4];
  const int* dpt = (const int*)d_in[5];
  const int* stp = (const int*)d_in[6];
  const int* dtp = (const int*)d_in[7];

  // params flattened in jax pytree (sorted dict-key) order per layer:
  // W/p, W/t, a_dst/pp, a_dst/pt, a_dst/tp, a_src/pp, a_src/pt, a_src/tp, b/p, b/t, kW, kb, q
  int i = 8;
  const float* l1_Wp    = (const float*)d_in[i++];
  const float* l1_Wt    = (const float*)d_in[i++];
  const float* l1_ad_pp = (const float*)d_in[i++];
  const float* l1_ad_pt = (const float*)d_in[i++];
  const float* l1_ad_tp = (const float*)d_in[i++];
  const float* l1_as_pp = (const float*)d_in[i++];
  const float* l1_as_pt = (const float*)d_in[i++];
  const float* l1_as_tp = (const float*)d_in[i++];
  const float* l1_bp    = (const float*)d_in[i++];
  const float* l1_bt    = (const float*)d_in[i++];
  const float* l1_kW    = (const float*)d_in[i++];
  const float* l1_kb    = (const float*)d_in[i++];
  const float* l1_q     = (const float*)d_in[i++];
  const float* l2_Wp    = (const float*)d_in[i++];
  const float* l2_Wt    = (const float*)d_in[i++];
  const float* l2_ad_pp = (const float*)d_in[i++];
  const float* l2_ad_pt = (const float*)d_in[i++];
  const float* l2_ad_tp = (const float*)d_in[i++];
  const float* l2_as_pp = (const float*)d_in[i++];
  const float* l2_as_pt = (const float*)d_in[i++];
  const float* l2_as_tp = (const float*)d_in[i++];
  const float* l2_bp    = (const float*)d_in[i++];
  const float* l2_bt    = (const float*)d_in[i++];
  const float* l2_kW    = (const float*)d_in[i++];
  const float* l2_kb    = (const float*)d_in[i++];
  const float* l2_q     = (const float*)d_in[i++];

  // workspace carve-out
  float* W = (float*)d_ws;
  size_t o = 0;
  auto alloc = [&](size_t n) { float* p = W + o; o += n; return p; };
  float* h_p   = alloc((size_t)NP * F1);
  float* h_t   = alloc((size_t)NT * F1);
  float* o_pp  = alloc((size_t)NP * F1);
  float* o_pt  = alloc((size_t)NT * F1);   // becomes h1_t after layer 1
  float* o_tp  = alloc((size_t)NP * F1);
  float* h1p   = alloc((size_t)NP * F1);
  float* h2p   = alloc((size_t)NP * F2);
  float* h2t   = alloc((size_t)NT * F2);
  float* o2_pp = alloc((size_t)NP * F2);
  float* o2_pt = alloc((size_t)NT * F2);
  float* o2_tp = alloc((size_t)NP * F2);
  float* as_b  = alloc((size_t)NP * 4);
  float* ad_b  = alloc((size_t)NP * 4);
  unsigned* m_b = (unsigned*)alloc((size_t)NP * 4);
  float* den_b = alloc((size_t)NP * 4);
  float* e_att = alloc((size_t)300000 * 4);
  float* wsum  = alloc(8);

  // ---- layer 1: heads=4, d=64, K=128 ----
  hanconv_layer(x_p, x_t, 128, 4, 64,
                spp, dpp, spt, dpt, stp, dtp,
                l1_Wp, l1_Wt, l1_bp, l1_bt,
                l1_as_pp, l1_as_pt, l1_as_tp,
                l1_ad_pp, l1_ad_pt, l1_ad_tp,
                l1_kW, l1_kb, l1_q,
                h_p, h_t, o_pp, o_pt, o_tp,
                as_b, ad_b, m_b, den_b, e_att, wsum,
                h1p, /*compute_t=*/true, stream);

  // ---- layer 2: heads=1, d=64, K=256; only "p" output needed -> skip pt relation ----
  hanconv_layer(h1p, o_pt, 256, 1, 64,
                spp, dpp, spt, dpt, stp, dtp,
                l2_Wp, l2_Wt, l2_bp, l2_bt,
                l2_as_pp, l2_as_pt, l2_as_tp,
                l2_ad_pp, l2_ad_pt, l2_ad_tp,
                l2_kW, l2_kb, l2_q,
                h2p, h2t, o2_pp, o2_pt, o2_tp,
                as_b, ad_b, m_b, den_b, e_att, wsum,
                (float*)d_out, /*compute_t=*/false, stream);
}